// scaled_dot_product_att_5746666242379
// MI455X (gfx1250) — compile-verified
//
#include <hip/hip_runtime.h>
#include <hip/hip_bf16.h>

typedef __attribute__((ext_vector_type(2))) float v2f;
typedef __attribute__((ext_vector_type(8))) float v8f;

namespace {

constexpr int S_LEN  = 512;   // sequence length
constexpr int BH     = 256;   // batch * heads
constexpr int D_HEAD = 64;
constexpr int HEADS  = 16;
constexpr int RTAB   = 2 * S_LEN - 1;  // 1023

constexpr int BM = 64;   // query rows per block (16 per wave, 4 waves)
constexpr int BN = 16;   // key cols per iteration
constexpr int NT = S_LEN / BN;  // 32 key tiles

constexpr int LDQ = 68;  // padded LDS row stride (floats): 16B-aligned rows, bank stagger

constexpr int KS_SZ = BN * LDQ;     // 1088 floats per K tile buffer
constexpr int VS_SZ = BN * D_HEAD;  // 1024 floats per V tile buffer

// LDS layout (float offsets)
constexpr int QS_OFF = 0;                        // [64][68]   Q rows of this block
constexpr int KQ_OFF = QS_OFF + BM * LDQ;        // [64][68]   K at the block's query rows (bias A operand)
constexpr int KS_OFF = KQ_OFF + BM * LDQ;        // 2x [16][68]  K tile, double buffered
constexpr int VS_OFF = KS_OFF + 2 * KS_SZ;       // 2x [16][64]  V tile, double buffered
constexpr int RS_OFF = VS_OFF + 2 * VS_SZ;       // [128][68]  circular rpos window (row = r & 127)
constexpr int RL_OFF = RS_OFF + 128 * LDQ;       // [4][16][32] per-wave rel-logits band
constexpr int PL_OFF = RL_OFF + 4 * 16 * 32;     // [4][16][18] per-wave P (C->A transpose staging)
constexpr int LDS_FLOATS = PL_OFF + 4 * 16 * 18; // 24832 floats = 99328 B

__device__ inline v8f wmma_f32x4(v2f a, v2f b, v8f c) {
  // D = A(16x4, f32) * B(4x16, f32) + C(16x16, f32)
  return __builtin_amdgcn_wmma_f32_16x16x4_f32(false, a, false, b, (short)0, c,
                                               false, false);
}

// 16B per-lane async copy: global -> LDS, tracked by ASYNCcnt (no VGPR staging).
__device__ inline void async_copy16(unsigned lds_byte_off, const float* gsrc) {
  asm volatile("global_load_async_to_lds_b128 %0, %1, off"
               :: "v"(lds_byte_off), "v"((unsigned long long)gsrc)
               : "memory");
}

__device__ inline void wait_async0() {
  asm volatile("s_wait_asynccnt 0x0" ::: "memory");
}

__global__ __launch_bounds__(128)
void rel_attn_kernel(const float* __restrict__ q, const float* __restrict__ k,
                     const float* __restrict__ v, const float* __restrict__ rpos,
                     float* __restrict__ out) {
  extern __shared__ float smem[];

  const int tid  = threadIdx.x;
  const int w    = tid >> 5;        // wave id 0..3
  const int lane = tid & 31;
  const int half = lane >> 4;       // 0: lanes 0-15, 1: lanes 16-31
  const int jl   = lane & 15;       // column / row-in-A index

  const int bh   = blockIdx.x & (BH - 1);
  const int iblk = blockIdx.x >> 8;
  const int i0   = iblk * BM;       // first query row of this block
  const int h    = bh & (HEADS - 1);
  const int i0w  = i0 + 16 * w;     // first query row of this wave

  // ---- async issue helpers (all-lane, uniform trip counts) ----
  auto issue_kv = [&](int t, int buf) {
    const int j0 = t * BN;
    for (int tt = tid; tt < BN * 16; tt += 128) {
      const int row = tt >> 4;
      const int c4  = (tt & 15) << 2;
      const size_t g = ((size_t)(j0 + row) * BH + bh) * D_HEAD + c4;
      async_copy16((unsigned)(KS_OFF + buf * KS_SZ + row * LDQ + c4) * 4u, k + g);
      async_copy16((unsigned)(VS_OFF + buf * VS_SZ + row * D_HEAD + c4) * 4u, v + g);
    }
  };
  auto issue_r = [&](int rfirst, int nrows) {
    for (int tt = tid; tt < nrows * 16; tt += 128) {
      const int idx = tt >> 4;
      const int c4  = (tt & 15) << 2;
      const int r   = rfirst + idx;          // >= 0 by construction
      const int rb  = r & 127;               // circular slot
      const int rc  = (r > RTAB - 1) ? (RTAB - 1) : r;  // clamp unused overshoot
      async_copy16((unsigned)(RS_OFF + rb * LDQ + c4) * 4u,
                   rpos + ((size_t)h * RTAB + rc) * D_HEAD + c4);
    }
  };

  // ---- prologue: Q rows, K-at-query-rows, tile 0 K/V, initial 80-row R window ----
  for (int t = tid; t < BM * 16; t += 128) {
    const int row = t >> 4;
    const int c4  = (t & 15) << 2;
    const size_t g = ((size_t)(i0 + row) * BH + bh) * D_HEAD + c4;
    async_copy16((unsigned)(QS_OFF + row * LDQ + c4) * 4u, q + g);
    async_copy16((unsigned)(KQ_OFF + row * LDQ + c4) * 4u, k + g);
  }
  issue_kv(0, 0);
  issue_r(448 - i0, 80);   // rows r0b(0)..r0b(0)+79, r0b(t) = 16t - i0 + 448

  // ---- online-softmax state: one row per accumulator VGPR ----
  float mrow[8], lrow[8];
  v8f o[4];
  const v8f vzero = {0.f, 0.f, 0.f, 0.f, 0.f, 0.f, 0.f, 0.f};
#pragma unroll
  for (int vv = 0; vv < 8; ++vv) { mrow[vv] = -1.0e30f; lrow[vv] = 0.0f; }
#pragma unroll
  for (int n = 0; n < 4; ++n) o[n] = vzero;

  for (int t = 0; t < NT; ++t) {
    wait_async0();      // own async copies for tile t have landed
    __syncthreads();    // everyone's copies visible; previous tile fully consumed

    // software pipeline: launch tile t+1 staging, then compute tile t
    if (t + 1 < NT) {
      issue_kv(t + 1, (t + 1) & 1);
      issue_r(16 * (t + 1) - i0 + 512, 16);  // 16 fresh rows sliding the window
    }

    const int j0  = t * BN;
    const int buf = t & 1;

    // ---- S = Q * K^T  (16x16, K-dim 64 as 16 chained x4 WMMAs) ----
    v8f sc = vzero;
    const int arow_q  = QS_OFF + (16 * w + jl) * LDQ + 2 * half;  // A: lane=M, half picks K pair
    const int arow_kq = KQ_OFF + (16 * w + jl) * LDQ + 2 * half;
    const int brow_k  = KS_OFF + buf * KS_SZ + jl * LDQ + 2 * half;
#pragma unroll
    for (int kb = 0; kb < D_HEAD; kb += 4) {
      v2f a = *(const v2f*)(smem + arow_q + kb);
      v2f b = *(const v2f*)(smem + brow_k + kb);
      sc = wmma_f32x4(a, b, sc);
    }

    // ---- rel-logits band RL = K[i-rows] * Rwin^T  (16x32, two N tiles) ----
    // wave band start r0w = j0 - i0 - 16w + 496; circular row = (r & 127)
    v8f rl0 = vzero, rl1 = vzero;
    const int rb0 = ((j0 - i0 + 496 - 16 * w) & 127) + jl;        // n = 0 cols
    const int rb1 = ((j0 - i0 + 496 - 16 * w + 16) & 127) + jl;   // n = 1 cols
    const int brow_r0 = RS_OFF + rb0 * LDQ + 2 * half;
    const int brow_r1 = RS_OFF + rb1 * LDQ + 2 * half;
#pragma unroll
    for (int kb = 0; kb < D_HEAD; kb += 4) {
      v2f a  = *(const v2f*)(smem + arow_kq + kb);
      v2f b0 = *(const v2f*)(smem + brow_r0 + kb);
      v2f b1 = *(const v2f*)(smem + brow_r1 + kb);
      rl0 = wmma_f32x4(a, b0, rl0);
      rl1 = wmma_f32x4(a, b1, rl1);
    }
    // spill RL band to per-wave LDS for the diagonal gather
#pragma unroll
    for (int vv = 0; vv < 8; ++vv) {
      const int M = vv + 8 * half;
      smem[RL_OFF + w * 512 + M * 32 + jl]      = rl0[vv];
      smem[RL_OFF + w * 512 + M * 32 + 16 + jl] = rl1[vv];
    }

    // ---- bias gather + scale + online softmax (rows live per-VGPR) ----
#pragma unroll
    for (int vv = 0; vv < 8; ++vv) {
      const int il = vv + 8 * half;                 // local query row
      float sv = (sc[vv] +
                  smem[RL_OFF + w * 512 + il * 32 + (jl - il + 15)]) * 0.125f;
      float mx = sv;                                // row max over 16 lanes of half
      mx = fmaxf(mx, __shfl_xor(mx, 1, 32));
      mx = fmaxf(mx, __shfl_xor(mx, 2, 32));
      mx = fmaxf(mx, __shfl_xor(mx, 4, 32));
      mx = fmaxf(mx, __shfl_xor(mx, 8, 32));
      const float mnew = fmaxf(mrow[vv], mx);
      const float corr = __expf(mrow[vv] - mnew);
      const float p    = __expf(sv - mnew);
      float rs = p;                                 // row sum
      rs += __shfl_xor(rs, 1, 32);
      rs += __shfl_xor(rs, 2, 32);
      rs += __shfl_xor(rs, 4, 32);
      rs += __shfl_xor(rs, 8, 32);
      lrow[vv] = lrow[vv] * corr + rs;
      mrow[vv] = mnew;
#pragma unroll
      for (int n = 0; n < 4; ++n) o[n][vv] *= corr;
      smem[PL_OFF + w * 288 + il * 18 + jl] = p;    // C-layout -> LDS (A layout reload)
    }

    // ---- O += P * V  (16x64, K-dim 16 as 4 chained x4 WMMAs per N tile) ----
#pragma unroll
    for (int kb = 0; kb < BN; kb += 4) {
      v2f a = *(const v2f*)(smem + PL_OFF + w * 288 + jl * 18 + kb + 2 * half);
#pragma unroll
      for (int n = 0; n < 4; ++n) {
        v2f b;
        b.x = smem[VS_OFF + buf * VS_SZ + (kb + 2 * half)     * D_HEAD + 16 * n + jl];
        b.y = smem[VS_OFF + buf * VS_SZ + (kb + 2 * half + 1) * D_HEAD + 16 * n + jl];
        o[n] = wmma_f32x4(a, b, o[n]);
      }
    }
  }

  // ---- epilogue: normalize by l and store [S, BH, D] ----
#pragma unroll
  for (int vv = 0; vv < 8; ++vv) {
    const float inv = 1.0f / lrow[vv];
    const int i = i0w + vv + 8 * half;
    const size_t base = ((size_t)i * BH + bh) * D_HEAD;
#pragma unroll
    for (int n = 0; n < 4; ++n)
      out[base + 16 * n + jl] = o[n][vv] * inv;
  }
}

}  // namespace

extern "C" void kernel_launch(void* const* d_in, const int* in_sizes, int n_in,
                              void* d_out, int out_size, void* d_ws, size_t ws_size,
                              hipStream_t stream) {
  const float* q    = (const float*)d_in[0];
  const float* k    = (const float*)d_in[1];
  const float* v    = (const float*)d_in[2];
  const float* rpos = (const float*)d_in[3];
  float* out        = (float*)d_out;

  const dim3 grid(BH * (S_LEN / BM));  // 256 bh * 8 row-blocks = 2048 WGs
  const dim3 block(128);               // 4 waves (wave32)
  const size_t smem_bytes = (size_t)LDS_FLOATS * sizeof(float);  // ~97 KB

  hipLaunchKernelGGL(rel_attn_kernel, grid, block, smem_bytes, stream,
                     q, k, v, rpos, out);
}